// LabelSmoothingRankingLoss_83227876262472
// MI455X (gfx1250) — compile-verified
//
#include <hip/hip_runtime.h>
#include <math.h>

typedef float v2f __attribute__((ext_vector_type(2)));
typedef float v8f __attribute__((ext_vector_type(8)));

#define MARGIN_F 1.0f
#define SMOOTH_F 0.1f
#define NEG_BIG  (-1.0e30f)
#define POS_BIG  (3.0e38f)

__device__ __forceinline__ float relu_med3(float x) {
    // single v_med3_f32: median{x, 0, +BIG} == relu(x); avoids fmaxf canonicalize
    return __builtin_amdgcn_fmed3f(x, 0.0f, POS_BIG);
}

// ---------------- K1: per-block max & sum over scores -----------------------
__global__ void k_partials(const float* __restrict__ scores, int n,
                           float* __restrict__ pmax, float* __restrict__ psum) {
    __shared__ float smax[256];
    __shared__ float ssum[256];
    int t = threadIdx.x;
    int idx = blockIdx.x * 256 + t;
    float v = (idx < n) ? scores[idx] : NEG_BIG;
    float s = (idx < n) ? v : 0.0f;
    if (idx + 256 < n) __builtin_prefetch(&scores[idx + 256], 0, 0); // global_prefetch_b8
    smax[t] = v; ssum[t] = s;
    __syncthreads();
    for (int off = 128; off > 0; off >>= 1) {
        if (t < off) {
            smax[t] = fmaxf(smax[t], smax[t + off]);
            ssum[t] += ssum[t + off];
        }
        __syncthreads();
    }
    if (t == 0) { pmax[blockIdx.x] = smax[0]; psum[blockIdx.x] = ssum[0]; }
}

// ---------------- K2: finish max/sum, gather GT scores ----------------------
__global__ void k_reduce1(const float* __restrict__ scores,
                          const int* __restrict__ gtpos,
                          int n, int g, int nb1,
                          float* __restrict__ stats,   // [0]=max [1]=sumAll [2]=sumGt [3]=gtCorr
                          float* __restrict__ gts,     // [g]
                          const float* __restrict__ pmax,
                          const float* __restrict__ psum) {
    __shared__ float smax[256];
    __shared__ float ssum[256];
    int t = threadIdx.x;
    float m = NEG_BIG, s = 0.0f;
    for (int i = t; i < nb1; i += 256) { m = fmaxf(m, pmax[i]); s += psum[i]; }
    smax[t] = m; ssum[t] = s;
    __syncthreads();
    for (int off = 128; off > 0; off >>= 1) {
        if (t < off) { smax[t] = fmaxf(smax[t], smax[t + off]); ssum[t] += ssum[t + off]; }
        __syncthreads();
    }
    if (t == 0) { stats[0] = smax[0]; stats[1] = ssum[0]; }
    float gs = 0.0f;
    for (int j = t; j < g; j += 256) { float v = scores[gtpos[j]]; gts[j] = v; gs += v; }
    __syncthreads();
    ssum[t] = gs;
    __syncthreads();
    for (int off = 128; off > 0; off >>= 1) {
        if (t < off) ssum[t] += ssum[t + off];
        __syncthreads();
    }
    if (t == 0) stats[2] = ssum[0];
}

// ---------------- K3: GT x GT correction (subtracted in finalize) -----------
__global__ void k_gtcorr(const float* __restrict__ gts, int g,
                         float* __restrict__ stats) {
    __shared__ float ldsGt[1024];
    __shared__ float red[256];
    int t = threadIdx.x;
    for (int j = t; j < g; j += 256) ldsGt[j] = gts[j];
    __syncthreads();
    float acc = 0.0f;
    for (int i = t; i < g; i += 256) {
        float si = ldsGt[i] + MARGIN_F;
        for (int j = 0; j < g; ++j) acc += relu_med3(si - ldsGt[j]);
    }
    red[t] = acc;
    __syncthreads();
    for (int off = 128; off > 0; off >>= 1) {
        if (t < off) red[t] += red[t + off];
        __syncthreads();
    }
    if (t == 0) stats[3] = red[0];
}

// ---------------- K4: exp partials + WMMA pairwise margin -------------------
// Block covers 256 scores. Wave w owns two 16-score strips (2x A, 2x acc).
// B rows are pre-staged in LDS in final per-lane layout: bneg[tile*32 + lane],
// lanes 16..31 pre-zeroed, gt pre-negated -> inner loop = 1 ds_load + 2 wmma.
__global__ void k_main(const float* __restrict__ scores, int n, int g,
                       const float* __restrict__ stats,
                       const float* __restrict__ gts,
                       float* __restrict__ expPartial,
                       float* __restrict__ marginPartial) {
    __shared__ float bneg[2048];   // up to 64 tiles * 32 lanes (g <= 1024)
    __shared__ float red[256];
    __shared__ float wsum[8];
    int t   = threadIdx.x;
    int blk = blockIdx.x;
    int ntiles = g >> 4;

    // stage B operand in per-lane layout
    for (int j = t; j < (ntiles << 5); j += 256) {
        int l = j & 31;
        bneg[j] = (l < 16) ? (-gts[((j >> 5) << 4) + l]) : 0.0f;
    }
    __syncthreads();

    // ---- exp(s - max) partial over this block's 256 scores ----
    float M = stats[0];
    int idx = blk * 256 + t;
    float e = (idx < n) ? __expf(scores[idx] - M) : 0.0f;
    red[t] = e;
    __syncthreads();
    for (int off = 128; off > 0; off >>= 1) {
        if (t < off) red[t] += red[t + off];
        __syncthreads();
    }
    if (t == 0) expPartial[blk] = red[0];
    __syncthreads();

    // ---- WMMA pairwise relu margin ----
    int lane   = t & 31;
    int w      = t >> 5;
    int base_m = blk * 256 + w * 32;          // two strips: base_m, base_m+16
    float partial = 0.0f;
    if (base_m + 32 <= n) {                   // wave-uniform guard: EXEC stays full
        bool lo = (lane < 16);
        float sv0 = lo ? scores[base_m + lane]       : 0.0f;
        float sv1 = lo ? scores[base_m + 16 + lane]  : 0.0f;
        v2f A0, A1;
        A0.x = lo ? (sv0 + MARGIN_F) : 0.0f;  A0.y = lo ? 1.0f : 0.0f;
        A1.x = lo ? (sv1 + MARGIN_F) : 0.0f;  A1.y = lo ? 1.0f : 0.0f;
        float bx = lo ? 1.0f : 0.0f;          // loop-invariant K=0 row of B
        v8f acc0 = {0.f,0.f,0.f,0.f,0.f,0.f,0.f,0.f};
        v8f acc1 = {0.f,0.f,0.f,0.f,0.f,0.f,0.f,0.f};
        const v8f czero = {0.f,0.f,0.f,0.f,0.f,0.f,0.f,0.f};
        for (int k = 0; k < ntiles; ++k) {
            v2f B;
            B.x = bx;
            B.y = bneg[(k << 5) + lane];      // unconditional ds_load_b32
            v8f d0 = __builtin_amdgcn_wmma_f32_16x16x4_f32(
                false, A0, false, B, (short)0, czero, false, false);
            v8f d1 = __builtin_amdgcn_wmma_f32_16x16x4_f32(
                false, A1, false, B, (short)0, czero, false, false);
#pragma unroll
            for (int c = 0; c < 8; ++c) {
                acc0[c] += relu_med3(d0[c]);
                acc1[c] += relu_med3(d1[c]);
            }
        }
        float ls = 0.0f;
#pragma unroll
        for (int c = 0; c < 8; ++c) ls += acc0[c] + acc1[c];
        for (int off = 16; off > 0; off >>= 1) ls += __shfl_down(ls, off, 32);
        partial = ls;
    }
    if (lane == 0) wsum[w] = partial;
    __syncthreads();
    if (t == 0) {
        float m2 = 0.0f;
        for (int i = 0; i < 8; ++i) m2 += wsum[i];
        marginPartial[blk] = m2;
    }
}

// ---------------- K5: finalize scalar ---------------------------------------
__global__ void k_final(int n, int g, int nb3,
                        const float* __restrict__ stats,
                        const float* __restrict__ expPartial,
                        const float* __restrict__ marginPartial,
                        float* __restrict__ out) {
    __shared__ float r1[256];
    __shared__ float r2[256];
    int t = threadIdx.x;
    float es = 0.0f, ms = 0.0f;
    for (int i = t; i < nb3; i += 256) { es += expPartial[i]; ms += marginPartial[i]; }
    r1[t] = es; r2[t] = ms;
    __syncthreads();
    for (int off = 128; off > 0; off >>= 1) {
        if (t < off) { r1[t] += r1[t + off]; r2[t] += r2[t + off]; }
        __syncthreads();
    }
    if (t == 0) {
        float M = stats[0], Sall = stats[1], Sgt = stats[2], corr = stats[3];
        float lse  = M + logf(r1[0]);
        float base = SMOOTH_F / (float)(n - g);
        float ce = lse - base * (Sall - Sgt) - ((1.0f - SMOOTH_F) / (float)g) * Sgt;
        float margin = (r2[0] - corr) / ((float)g * (float)(n - g));
        out[0] = ce + 0.5f * margin;
    }
}

extern "C" void kernel_launch(void* const* d_in, const int* in_sizes, int n_in,
                              void* d_out, int out_size, void* d_ws, size_t ws_size,
                              hipStream_t stream) {
    (void)n_in; (void)out_size; (void)ws_size;
    const float* scores = (const float*)d_in[0];
    const int*   gtpos  = (const int*)d_in[1];
    int n = in_sizes[0];
    int g = in_sizes[1];

    float* ws    = (float*)d_ws;
    float* stats = ws;            // 4 floats
    float* gts   = ws + 4;        // g floats
    int nb1 = (n + 255) / 256;
    int nb3 = (n + 255) / 256;    // k_main blocks (256 scores each)
    float* pmax = gts + g;
    float* psum = pmax + nb1;
    float* expP = psum + nb1;
    float* marP = expP + nb3;
    float* out  = (float*)d_out;

    k_partials<<<nb1, 256, 0, stream>>>(scores, n, pmax, psum);
    k_reduce1 <<<1,   256, 0, stream>>>(scores, gtpos, n, g, nb1, stats, gts, pmax, psum);
    k_gtcorr  <<<1,   256, 0, stream>>>(gts, g, stats);
    k_main    <<<nb3, 256, 0, stream>>>(scores, n, g, stats, gts, expP, marP);
    k_final   <<<1,   256, 0, stream>>>(n, g, nb3, stats, expP, marP, out);
}